// Discounted_Cumsum_25374666785435
// MI455X (gfx1250) — compile-verified
//
#include <hip/hip_runtime.h>

typedef __attribute__((ext_vector_type(2))) float v2f;
typedef __attribute__((ext_vector_type(8))) float v8f;

#define S_LEN 4096
#define D_LEN 64
#define CHUNKS (S_LEN / 16)

// One wave32 per (b, h, 16-wide feature tile). In-chunk scan of 16 timesteps is
// a 16x16 (lower-triangular gamma-power matrix) x 16x16 feature tile matmul done
// with 4 accumulating V_WMMA_F32_16X16X4_F32 ops; cross-chunk carry folded into
// the accumulator init and extracted from row 15 of the C/D layout by a shuffle.
__global__ __launch_bounds__(256) void dcumsum_wmma(const float* __restrict__ x,
                                                    const float* __restrict__ g,
                                                    float* __restrict__ y)
{
    const int lane = threadIdx.x & 31;
    const int wid  = blockIdx.x * (blockDim.x >> 5) + (threadIdx.x >> 5); // 0..511
    const int nt   = wid & 3;        // feature tile (D/16)
    const int bh   = wid >> 2;       // b*16 + h
    const int h    = bh & 15;

    const int half = lane >> 4;      // which half-wave
    const int lm   = lane & 15;      // M (A-frag) / N (B,C,D frags)

    const float gamma = g[h];

    // small integer power (exponent <= 16), only used at setup
    auto ipow = [&](int e) -> float {
        float r = 1.0f;
        for (int i = 0; i < e; ++i) r *= gamma;
        return r;
    };

    // A fragments: A[m][k] = gamma^(m-k) for m>=k else 0.  16x4 f32 layout:
    // lanes 0-15 hold K={0,1} in the 2 VGPRs, lanes 16-31 hold K={2,3}.
    v2f afrag[4];
#pragma unroll
    for (int kb = 0; kb < 4; ++kb) {
        const int k0 = kb * 4 + half * 2;
        afrag[kb].x = (lm >= k0)     ? ipow(lm - k0)     : 0.0f;
        afrag[kb].y = (lm >= k0 + 1) ? ipow(lm - k0 - 1) : 0.0f;
    }

    // carry scale for C/D VGPR r: row (r + half*8) gets gamma^(row+1)
    float cs[8];
    cs[0] = half ? ipow(9) : gamma;
#pragma unroll
    for (int r = 1; r < 8; ++r) cs[r] = cs[r - 1] * gamma;

    const float* xb = x + (size_t)bh * S_LEN * D_LEN + nt * 16 + lm;
    float*       yb = y + (size_t)bh * S_LEN * D_LEN + nt * 16 + lm;

    float carry = 0.0f;
    float bufA[8], bufB[8];

    // B fragment for chunk c: B[k][n], n = lm; lanes 0-15 take K={k0,k0+1},
    // lanes 16-31 take K={k0+2,k0+3}  (mirrors the A-frag K striping)
    auto load_chunk = [&](int c, float* bf) {
        const int row0 = c * 16 + half * 2;
#pragma unroll
        for (int kb = 0; kb < 4; ++kb) {
            bf[2 * kb + 0] = xb[(size_t)(row0 + kb * 4 + 0) * D_LEN];
            bf[2 * kb + 1] = xb[(size_t)(row0 + kb * 4 + 1) * D_LEN];
        }
    };

    auto compute_store = [&](int c, const float* bf) {
        v8f acc;
#pragma unroll
        for (int r = 0; r < 8; ++r) acc[r] = cs[r] * carry; // gamma^(i+1)*carry
#pragma unroll
        for (int kb = 0; kb < 4; ++kb) {
            v2f bm;
            bm.x = bf[2 * kb + 0];
            bm.y = bf[2 * kb + 1];
            acc = __builtin_amdgcn_wmma_f32_16x16x4_f32(
                false, afrag[kb], false, bm, (short)0, acc, false, false);
        }
        // new carry = y[row 15][n]; row 15 lives in VGPR 7, lanes 16-31
        carry = __shfl(acc[7], 16 + lm, 32);
#pragma unroll
        for (int r = 0; r < 8; ++r)
            yb[(size_t)(c * 16 + r + half * 8) * D_LEN] = acc[r];
    };

    // software-pipelined: next chunk's loads are in flight during WMMA chain
    load_chunk(0, bufA);
    for (int c = 0; c < CHUNKS; c += 2) {
        load_chunk(c + 1, bufB);
        compute_store(c, bufA);
        if (c + 2 < CHUNKS) load_chunk(c + 2, bufA);
        compute_store(c + 1, bufB);
    }
}

extern "C" void kernel_launch(void* const* d_in, const int* in_sizes, int n_in,
                              void* d_out, int out_size, void* d_ws, size_t ws_size,
                              hipStream_t stream)
{
    const float* x = (const float*)d_in[0]; // (8,16,4096,64) f32
    const float* g = (const float*)d_in[1]; // (16,) f32
    float* y = (float*)d_out;

    // 512 waves: one per (b,h,feature-tile); 8 waves per 256-thread block
    dim3 grid(64), block(256);
    dcumsum_wmma<<<grid, block, 0, stream>>>(x, g, y);
}